// SpatialClusterPruning_70093866270896
// MI455X (gfx1250) — compile-verified
//
#include <hip/hip_runtime.h>
#include <stdint.h>

// Problem constants (from the reference: CELL=32, IMG=4096, B=8)
#define BATCH   8
#define CELL_INV 0.03125f       // 1/32, exact power of two
#define NCOLS   130             // IMG/CELL + 2
#define NCELLS  (130 * 130)     // 16900
#define TILE    4096            // elements per block tile
#define TPB     256             // threads per block
#define IPT     (TILE / TPB)    // 16 items per thread (blocked -> stable compaction)

typedef __attribute__((ext_vector_type(4))) float f4;
typedef __attribute__((ext_vector_type(2))) float f2;

// ---------------------------------------------------------------------------
// keep test: cell key -> compare against per-(batch,cell) max score
// scores are in [0,1) so uint bit-compare == float compare for the atomic max.
__device__ __forceinline__ bool keep_of(float x, float y, float s,
                                        const unsigned* __restrict__ cm) {
    int cx = (int)(x * CELL_INV);       // trunc == floor for x >= 0
    int cy = (int)(y * CELL_INV);
    return s >= __uint_as_float(cm[cy * NCOLS + cx]);
}

// Stage one TILE of scores (16 KB) + keypoints (32 KB) into LDS via the
// gfx1250 async-to-LDS path (ASYNCcnt). Sources are clamped to the batch
// region so the last partial tile never reads OOB. Caller must __syncthreads()
// after this returns (each wave has already waited on its own ASYNCcnt).
__device__ __forceinline__ void stage_tile_async(const char* scB, const char* kpB,
                                                 float* s_sc, float* s_kp,
                                                 long tile0, int N, unsigned t) {
    unsigned scMax = (unsigned)((size_t)N * 4 - 16);
    unsigned kpMax = (unsigned)((size_t)N * 8 - 16);
    unsigned ldsS = (unsigned)(uintptr_t)s_sc;   // addr[31:0] == LDS byte address
    unsigned ldsK = (unsigned)(uintptr_t)s_kp;
    // scores tile: TILE*4 B = 1024 x 16B chunks, 4 per thread
#pragma unroll
    for (int k = 0; k < 4; ++k) {
        unsigned c = t + (unsigned)k * TPB;
        unsigned dst = ldsS + c * 16u;
        unsigned src = (unsigned)(tile0 * 4) + c * 16u;
        if (src > scMax) src = scMax;
        asm volatile("global_load_async_to_lds_b128 %0, %1, %2"
                     :: "v"(dst), "v"(src), "s"(scB) : "memory");
    }
    // keypoints tile: TILE*8 B = 2048 x 16B chunks, 8 per thread
#pragma unroll
    for (int k = 0; k < 8; ++k) {
        unsigned c = t + (unsigned)k * TPB;
        unsigned dst = ldsK + c * 16u;
        unsigned src = (unsigned)(tile0 * 8) + c * 16u;
        if (src > kpMax) src = kpMax;
        asm volatile("global_load_async_to_lds_b128 %0, %1, %2"
                     :: "v"(dst), "v"(src), "s"(kpB) : "memory");
    }
    asm volatile("s_wait_asynccnt 0x0" ::: "memory");
}

// ---------------------------------------------------------------------------
__global__ void k_init_cellmax(unsigned* __restrict__ cellMax, int total) {
    int i = blockIdx.x * blockDim.x + threadIdx.x;
    if (i < total) cellMax[i] = 0u;     // bits(0.0f); scores >= 0 so this is a safe floor
}

__global__ void k_zero_f4(f4* __restrict__ p, int n4) {
    int i = blockIdx.x * blockDim.x + threadIdx.x;
    if (i < n4) {
        f4 z = {0.f, 0.f, 0.f, 0.f};
        __builtin_nontemporal_store(z, p + i);   // NT: don't evict inputs from L2
    }
}

// ---------------------------------------------------------------------------
// Pass 1: per-(batch,cell) max score via L2 atomics. 4 elements per thread,
// b128 loads of both keypoints and scores (regular loads: first touch fills L2).
__global__ __launch_bounds__(TPB)
void k_cell_atomic_max(const float* __restrict__ kp, const float* __restrict__ sc,
                       unsigned* __restrict__ cellMax, int N) {
    long t = (long)blockIdx.x * blockDim.x + threadIdx.x;
    long i0 = t * 4;
    if (i0 >= (long)BATCH * N) return;          // N % 4 == 0, no straddle
    int b = (int)(i0 / N);
    unsigned* cm = cellMax + b * NCELLS;
    f4 s4 = *(const f4*)(sc + i0);
    f4 k0 = *(const f4*)(kp + 2 * i0);      // kpt0.xy, kpt1.xy
    f4 k1 = *(const f4*)(kp + 2 * i0 + 4);  // kpt2.xy, kpt3.xy
    {
        int key = (int)(k0.y * CELL_INV) * NCOLS + (int)(k0.x * CELL_INV);
        atomicMax(&cm[key], __float_as_uint(s4.x));
    }
    {
        int key = (int)(k0.w * CELL_INV) * NCOLS + (int)(k0.z * CELL_INV);
        atomicMax(&cm[key], __float_as_uint(s4.y));
    }
    {
        int key = (int)(k1.y * CELL_INV) * NCOLS + (int)(k1.x * CELL_INV);
        atomicMax(&cm[key], __float_as_uint(s4.z));
    }
    {
        int key = (int)(k1.w * CELL_INV) * NCOLS + (int)(k1.z * CELL_INV);
        atomicMax(&cm[key], __float_as_uint(s4.w));
    }
}

// ---------------------------------------------------------------------------
// Pass 2: async-stage the tile to LDS, write keep mask (0.0/1.0 float, NT
// b128 stores) and per-block kept counts.
__global__ __launch_bounds__(TPB)
void k_mask_count(const float* __restrict__ kp, const float* __restrict__ sc,
                  const unsigned* __restrict__ cellMax,
                  float* __restrict__ outMask, unsigned* __restrict__ blockCounts,
                  int N, int nblk) {
    __shared__ float s_sc[TILE];          // 16 KB
    __shared__ float s_kp[TILE * 2];      // 32 KB
    __shared__ unsigned s_cnt;

    int b   = blockIdx.x / nblk;
    int blk = blockIdx.x % nblk;
    long tile0 = (long)blk * TILE;
    unsigned t = threadIdx.x;

    stage_tile_async((const char*)(sc + (size_t)b * N),
                     (const char*)(kp + (size_t)b * N * 2),
                     s_sc, s_kp, tile0, N, t);
    if (t == 0) s_cnt = 0;
    __syncthreads();

    int loc0 = (int)t * IPT;
    long gi0 = tile0 + loc0;
    unsigned cnt = 0;
    if (gi0 < N) {   // N % IPT == 0 -> a thread is fully in or fully out
        float* mB = outMask + (size_t)b * N;
        const unsigned* cm = cellMax + b * NCELLS;
#pragma unroll
        for (int g = 0; g < IPT / 4; ++g) {
            int lj = loc0 + g * 4;
            f4 m;
#pragma unroll
            for (int e = 0; e < 4; ++e) {
                bool kpn = keep_of(s_kp[(lj + e) * 2], s_kp[(lj + e) * 2 + 1],
                                   s_sc[lj + e], cm);
                m[e] = kpn ? 1.f : 0.f;
                cnt += kpn;
            }
            __builtin_nontemporal_store(m, (f4*)(mB + gi0 + g * 4));
        }
    }
    int w = (int)cnt;
#pragma unroll
    for (int d = 16; d; d >>= 1) w += __shfl_down(w, d, 32);   // wave32 reduce
    if ((threadIdx.x & 31) == 0) atomicAdd(&s_cnt, (unsigned)w);
    __syncthreads();
    if (threadIdx.x == 0) blockCounts[(size_t)b * nblk + blk] = s_cnt;
}

// ---------------------------------------------------------------------------
// Pass 3: exclusive scan of block counts, one block per batch (nblk <= 512).
__global__ __launch_bounds__(512)
void k_scan(const unsigned* __restrict__ blockCounts,
            unsigned* __restrict__ blockOffsets, int nblk) {
    __shared__ unsigned lds[512];
    int b = blockIdx.x, t = threadIdx.x;
    unsigned v = (t < nblk) ? blockCounts[(size_t)b * nblk + t] : 0u;
    lds[t] = v;
    __syncthreads();
    for (int off = 1; off < 512; off <<= 1) {
        unsigned add = (t >= off) ? lds[t - off] : 0u;
        __syncthreads();
        lds[t] += add;
        __syncthreads();
    }
    if (t < nblk) blockOffsets[(size_t)b * nblk + t] = lds[t] - v;  // exclusive
}

// ---------------------------------------------------------------------------
// Pass 4: stable scatter. Tile DMA'd to LDS (async), counted, wave32-scanned,
// then scattered in original order with NT stores.
__global__ __launch_bounds__(TPB)
void k_scatter(const float* __restrict__ kp, const float* __restrict__ sc,
               const unsigned* __restrict__ cellMax,
               const unsigned* __restrict__ blockOffsets,
               float* __restrict__ outKpts, float* __restrict__ outScores,
               int N, int nblk) {
    __shared__ float s_sc[TILE];          // 16 KB
    __shared__ float s_kp[TILE * 2];      // 32 KB
    __shared__ unsigned s_wb[8];

    int b   = blockIdx.x / nblk;
    int blk = blockIdx.x % nblk;
    long tile0 = (long)blk * TILE;
    unsigned t = threadIdx.x;

    stage_tile_async((const char*)(sc + (size_t)b * N),
                     (const char*)(kp + (size_t)b * N * 2),
                     s_sc, s_kp, tile0, N, t);
    __syncthreads();

    // phase A: per-thread keep count over its 16 blocked items (stable order)
    const unsigned* cm = cellMax + b * NCELLS;
    int loc0 = (int)t * IPT;
    long gi0 = tile0 + loc0;
    unsigned cnt = 0, kmask = 0;
    if (gi0 < N) {
#pragma unroll
        for (int j = 0; j < IPT; ++j) {
            float x = s_kp[(loc0 + j) * 2];
            float y = s_kp[(loc0 + j) * 2 + 1];
            float s = s_sc[loc0 + j];
            if (keep_of(x, y, s, cm)) { kmask |= (1u << j); ++cnt; }
        }
    }

    // wave32 inclusive scan of per-thread counts
    int lane = (int)t & 31, warp = (int)t >> 5;
    int inc = (int)cnt;
#pragma unroll
    for (int d = 1; d < 32; d <<= 1) {
        int v = __shfl_up(inc, d, 32);
        if (lane >= d) inc += v;
    }
    if (lane == 31) s_wb[warp] = (unsigned)inc;
    __syncthreads();
    if (t == 0) {                        // tiny exclusive scan over 8 warp sums
        unsigned r = 0;
#pragma unroll
        for (int w = 0; w < 8; ++w) { unsigned x = s_wb[w]; s_wb[w] = r; r += x; }
    }
    __syncthreads();

    unsigned pos = blockOffsets[(size_t)b * nblk + blk] + s_wb[warp]
                 + (unsigned)inc - cnt;  // exclusive within wave

    // phase B: ordered scatter from LDS (NT stores: outputs are write-once)
    float* okp = outKpts  + (size_t)b * N * 2;
    float* osc = outScores + (size_t)b * N;
    if (gi0 < N) {
#pragma unroll
        for (int j = 0; j < IPT; ++j) {
            if (kmask & (1u << j)) {
                f2 xy = {s_kp[(loc0 + j) * 2], s_kp[(loc0 + j) * 2 + 1]};
                __builtin_nontemporal_store(xy, (f2*)(okp + (size_t)pos * 2));
                __builtin_nontemporal_store(s_sc[loc0 + j], osc + pos);
                ++pos;
            }
        }
    }
}

// ---------------------------------------------------------------------------
extern "C" void kernel_launch(void* const* d_in, const int* in_sizes, int n_in,
                              void* d_out, int out_size, void* d_ws, size_t ws_size,
                              hipStream_t stream) {
    const float* kp = (const float*)d_in[0];   // (B, N, 2)
    const float* sc = (const float*)d_in[1];   // (B, N)
    float* out = (float*)d_out;

    int BN = in_sizes[1];            // B * N
    int N  = BN / BATCH;             // 2,000,000 (multiple of IPT)
    int nblk = (N + TILE - 1) / TILE;   // 489 (<= 512, fits k_scan)

    unsigned* cellMax      = (unsigned*)d_ws;                       // B*NCELLS
    unsigned* blockCounts  = cellMax + (size_t)BATCH * NCELLS;      // B*nblk
    unsigned* blockOffsets = blockCounts + (size_t)BATCH * nblk;    // B*nblk

    float* outMask   = out;                     // (B,N)  mask as 0/1 float
    float* outKpts   = out + (size_t)BN;        // (B,N,2) zero-padded
    float* outScores = out + (size_t)BN * 3;    // (B,N)   zero-padded

    int totCells = BATCH * NCELLS;
    k_init_cellmax<<<(totCells + 255) / 256, 256, 0, stream>>>(cellMax, totCells);

    int zero4 = (BN * 3) / 4;                   // pruned kpts+scores region
    k_zero_f4<<<(zero4 + 255) / 256, 256, 0, stream>>>((f4*)outKpts, zero4);

    long tot4 = (long)BN / 4;
    k_cell_atomic_max<<<(int)((tot4 + TPB - 1) / TPB), TPB, 0, stream>>>(kp, sc, cellMax, N);

    k_mask_count<<<BATCH * nblk, TPB, 0, stream>>>(kp, sc, cellMax, outMask,
                                                   blockCounts, N, nblk);

    k_scan<<<BATCH, 512, 0, stream>>>(blockCounts, blockOffsets, nblk);

    k_scatter<<<BATCH * nblk, TPB, 0, stream>>>(kp, sc, cellMax, blockOffsets,
                                                outKpts, outScores, N, nblk);
}